// AdaAugment_1726576855582
// MI455X (gfx1250) — compile-verified
//
#include <hip/hip_runtime.h>
#include <math.h>

// ---------------------------------------------------------------------------
// AdaAugment pipeline on gfx1250:
//   reflect-pad(38) -> 2x sym6 upsample -> affine bilinear warp -> 2x sym6
//   downsample, N=32, C=3, 256x256 fp32.
// All FIR passes are expressed as fp32 WMMA (v_wmma_f32_16x16x4_f32) against
// on-the-fly Toeplitz filter matrices; only ONE intermediate (664x664,
// ~169 MB, fits the 192 MB L2) is materialized in d_ws.
// This revision makes all gathers BRANCHLESS (clamped address + v_cndmask on
// the value) so loads issue back-to-back instead of behind exec-mask branches.
// ---------------------------------------------------------------------------

typedef float v2f __attribute__((ext_vector_type(2)));
typedef float v8f __attribute__((ext_vector_type(8)));

#define NIMG 32
#define NCH  3
#define IMS  256     // source image H=W
#define MARG 38      // reflect margin (2*HZ_PAD + 32)
#define PADS 332     // 256 + 2*38
#define UPS  664     // 2*PADS  (upsampled intermediate)
#define WRP  524     // (256 + 2*HZ_PAD) * 2 (warp target)
#define OUTS 256

// D = A(16x4) * B(4x16) + C, all fp32 (wave32, 2 VGPRs per A/B, 8 per C/D)
__device__ __forceinline__ v8f wmma4(v2f a, v2f b, v8f c) {
  return __builtin_amdgcn_wmma_f32_16x16x4_f32(
      /*neg_a=*/false, a, /*neg_b=*/false, b,
      /*c_mod=*/(short)0, c, /*reuse_a=*/false, /*reuse_b=*/false);
}

// padded-space index -> source index in [0,256) ('reflect' mode, no edge dup)
// Valid (result in [0,256)) for any p in [-472, 548] -- covers every index we
// ever generate, so the load address is ALWAYS in-bounds.
__device__ __forceinline__ int reflect_idx(int p) {
  int r = p - MARG;
  if (r < 0) r = -r;
  if (r > IMS - 1) r = 2 * (IMS - 1) - r;
  return r;
}

// Polyphase 2x-upsample Toeplitz weight Wup[k][n] (16x16), gain 2 folded in.
// out[j0+n] = sum_k Wup[k][n] * in[j0/2 - 3 + k]   (zero outside [0,PADS))
__device__ __forceinline__ float wup(const float* fl, int k, int n) {
  int h = n >> 1;
  int b = ((n & 1) == 0) ? (h + 5 - k) : (h + 6 - k);
  bool ok = (b >= 0) && (b < 6);
  int bi = ok ? b : 0;
  float w = ((n & 1) == 0) ? fl[2 * bi + 1] : fl[2 * bi];
  return ok ? 2.0f * w : 0.0f;
}

// 2x-downsample Toeplitz weight Wdn[k][n] (48x16).
// out[j0+n] = sum_k Wdn[k][n] * in[2*j0 + k]  with Wdn[k][n] = f[k-2n-1]
__device__ __forceinline__ float wdn(const float* fl, int k, int n) {
  int t = k - 2 * n - 1;
  bool ok = (t >= 0) && (t < 12);
  float w = fl[ok ? t : 0];
  return ok ? w : 0.0f;
}

// ---------------------------------------------------------------------------
// K0: compose per-image inverse affine (2x3) exactly as the reference chain.
// ---------------------------------------------------------------------------
__global__ void affine_prep(const float* __restrict__ th,
                            const float* __restrict__ ls,
                            const float* __restrict__ tx,
                            const float* __restrict__ ty,
                            float* __restrict__ aff) {
  int i = threadIdx.x;
  if (i >= NIMG) return;
  float c = cosf(th[i]), sn = sinf(th[i]);
  float is = 1.0f / expf(ls[i]);
  float txp = -tx[i] * (float)IMS, typ = -ty[i] * (float)IMS;
  // scl_inv @ rot_inv @ trn_inv
  float g00 = c * is, g01 = sn * is;
  float g10 = -sn * is, g11 = c * is;
  float g02 = g00 * txp + g01 * typ;
  float g12 = g10 * txp + g11 * typ;
  // S2 @ G @ S0.5 : translation * 2
  g02 *= 2.0f; g12 *= 2.0f;
  // T(-.5) @ G @ T(.5)
  g02 += 0.5f * (g00 + g01) - 0.5f;
  g12 += 0.5f * (g10 + g11) - 0.5f;
  // diag(2/Wu,2/Hu,1) @ G @ diag(Wt/2,Ht/2,1)
  const float ri = 2.0f / (float)UPS;
  const float hw = 0.5f * (float)WRP;
  g00 *= ri * hw; g01 *= ri * hw; g02 *= ri;
  g10 *= ri * hw; g11 *= ri * hw; g12 *= ri;
  float* o = aff + i * 8;
  o[0] = g00; o[1] = g01; o[2] = g02;
  o[3] = g10; o[4] = g11; o[5] = g12;
  o[6] = 0.0f; o[7] = 0.0f;
}

// ---------------------------------------------------------------------------
// K1: fused reflect-pad + separable 2x upsample. One wave = one 16x16 tile of
// the 664x664 intermediate: tmp = Src x Wup  (4 WMMA), out = Wup^T x tmp (4).
// tmp crosses D-layout -> B-layout through a stride-17 LDS tile.
// ---------------------------------------------------------------------------
__global__ void __launch_bounds__(32) up2x(const float* __restrict__ img,
                                           const float* __restrict__ f,
                                           float* __restrict__ t2) {
  const int TPL = 42;  // ceil(664/16)
  int bid = blockIdx.x;
  int tc = bid % TPL; bid /= TPL;
  int tr = bid % TPL; int nc = bid / TPL;
  int r0 = tr * 16, j0 = tc * 16;
  int lane = threadIdx.x & 31;
  int half = lane >> 4, l = lane & 15;

  __shared__ float fl[12];
  __shared__ float tmp[16 * 17];
  if (lane < 12) fl[lane] = f[lane];
  __syncthreads();

  const float* src = img + (size_t)nc * IMS * IMS;
  float* dst = t2 + (size_t)nc * UPS * UPS;

  // ---- stage 1: horizontal upsample over a 16x16 source patch ----
  int rb = r0 / 2 - 3;            // padded-row base of the V-pass window
  int c0 = j0 / 2 - 3;            // padded-col base of the H-pass window
  int rp = rb + l;                // this lane's A row (padded space)
  bool rok = (rp >= 0) && (rp < PADS);   // V-pass zero-pads outside [0,PADS)
  const float* rowp = src + reflect_idx(rp) * IMS;

  v8f acc = {0.f, 0.f, 0.f, 0.f, 0.f, 0.f, 0.f, 0.f};
#pragma unroll
  for (int kk = 0; kk < 4; ++kk) {
    v2f a, b;
#pragma unroll
    for (int v = 0; v < 2; ++v) {
      int k = 4 * kk + 2 * half + v;
      int cp = c0 + k;
      // Branchless: address is always valid; mask the VALUE instead.
      float av = rowp[reflect_idx(cp)];
      bool cok = rok && (cp >= 0) && (cp < PADS);
      a[v] = cok ? av : 0.0f;     // A[r=l][k]
      b[v] = wup(fl, k, l);       // B[k][n=l]
    }
    acc = wmma4(a, b, acc);
  }
#pragma unroll
  for (int vi = 0; vi < 8; ++vi)  // D[M=vi+8*half][N=l] -> tmp[row][col]
    tmp[(vi + 8 * half) * 17 + l] = acc[vi];
  __syncthreads();

  // ---- stage 2: vertical upsample: out = Wup^T x tmp ----
  v8f o = {0.f, 0.f, 0.f, 0.f, 0.f, 0.f, 0.f, 0.f};
#pragma unroll
  for (int kk = 0; kk < 4; ++kk) {
    v2f a, b;
#pragma unroll
    for (int v = 0; v < 2; ++v) {
      int k = 4 * kk + 2 * half + v;
      a[v] = wup(fl, k, l);       // A[n=l][k] = Wup[k][n]
      b[v] = tmp[k * 17 + l];     // B[k][c=l]
    }
    o = wmma4(a, b, o);
  }
#pragma unroll
  for (int vi = 0; vi < 8; ++vi) {
    int orow = r0 + vi + 8 * half;
    int ocol = j0 + l;
    if (orow < UPS && ocol < UPS) dst[(size_t)orow * UPS + ocol] = o[vi];
  }
}

// Branchless bilinear tap: clamp coords (address always valid), mask value.
__device__ __forceinline__ float tex2d(const float* p, int y, int x) {
  int xc = min(max(x, 0), UPS - 1);
  int yc = min(max(y, 0), UPS - 1);
  float v = p[yc * UPS + xc];
  bool ok = ((unsigned)x < (unsigned)UPS) && ((unsigned)y < (unsigned)UPS);
  return ok ? v : 0.0f;
}

// ---------------------------------------------------------------------------
// K2: fused affine grid-sample + separable 2x downsample. One wave = one
// 16x16 tile of the final image. It bilinearly samples its 48x48 warped-space
// window into LDS (gathers hit L2: the 169 MB intermediate fits in 192 MB L2),
// then tmp(48x16) = Sg x Wdn (36 WMMA) and out = Wdn^T x tmp (12 WMMA).
// Rows/cols of the window with zero filter weight tolerate garbage samples.
// ---------------------------------------------------------------------------
__global__ void __launch_bounds__(32) warp_down(const float* __restrict__ t2,
                                               const float* __restrict__ f,
                                               const float* __restrict__ aff,
                                               float* __restrict__ out) {
  int bid = blockIdx.x;
  int tc = bid & 15; bid >>= 4;
  int tr = bid & 15; int nc = bid >> 4;
  int n_img = nc / NCH;
  int r0 = tr * 16, j0 = tc * 16;
  int lane = threadIdx.x & 31;
  int half = lane >> 4, l = lane & 15;

  __shared__ float fl[12];
  __shared__ float sg[48 * 49];
  __shared__ float tp[48 * 17];
  if (lane < 12) fl[lane] = f[lane];
  __syncthreads();

  const float* src = t2 + (size_t)nc * UPS * UPS;
  const float* A = aff + n_img * 8;
  float a00 = A[0], a01 = A[1], a02 = A[2];
  float a10 = A[3], a11 = A[4], a12 = A[5];

  // ---- bilinear grid-sample of the 48x48 window (72 samples/lane) ----
  for (int s = lane; s < 48 * 48; s += 32) {
    int ky = s / 48, kx = s % 48;
    float ox = (float)(2 * j0 + kx);
    float oy = (float)(2 * r0 + ky);
    float xs = (2.0f * ox + 1.0f) * (1.0f / (float)WRP) - 1.0f;
    float ys = (2.0f * oy + 1.0f) * (1.0f / (float)WRP) - 1.0f;
    float gx = a00 * xs + a01 * ys + a02;
    float gy = a10 * xs + a11 * ys + a12;
    float ix = (gx + 1.0f) * (0.5f * (float)UPS) - 0.5f;
    float iy = (gy + 1.0f) * (0.5f * (float)UPS) - 0.5f;
    float xf = floorf(ix), yf = floorf(iy);
    float wx = ix - xf, wy = iy - yf;
    int x0 = (int)xf, y0 = (int)yf;
    float v00 = tex2d(src, y0, x0);
    float v01 = tex2d(src, y0, x0 + 1);
    float v10 = tex2d(src, y0 + 1, x0);
    float v11 = tex2d(src, y0 + 1, x0 + 1);
    sg[ky * 49 + kx] = v00 * (1.0f - wx) * (1.0f - wy) + v01 * wx * (1.0f - wy)
                     + v10 * (1.0f - wx) * wy + v11 * wx * wy;
  }
  __syncthreads();

  // ---- stage 1: horizontal downsample (3 chunks of 16 rows, K = 48) ----
#pragma unroll
  for (int rc = 0; rc < 3; ++rc) {
    v8f acc = {0.f, 0.f, 0.f, 0.f, 0.f, 0.f, 0.f, 0.f};
#pragma unroll
    for (int kk = 0; kk < 12; ++kk) {
      v2f a, b;
#pragma unroll
      for (int v = 0; v < 2; ++v) {
        int k = 4 * kk + 2 * half + v;
        a[v] = sg[(rc * 16 + l) * 49 + k];  // A[r=l][k]
        b[v] = wdn(fl, k, l);               // B[k][n=l]
      }
      acc = wmma4(a, b, acc);
    }
#pragma unroll
    for (int vi = 0; vi < 8; ++vi)
      tp[(rc * 16 + vi + 8 * half) * 17 + l] = acc[vi];
  }
  __syncthreads();

  // ---- stage 2: vertical downsample: out = Wdn^T x tmp (K = 48) ----
  v8f o = {0.f, 0.f, 0.f, 0.f, 0.f, 0.f, 0.f, 0.f};
#pragma unroll
  for (int kk = 0; kk < 12; ++kk) {
    v2f a, b;
#pragma unroll
    for (int v = 0; v < 2; ++v) {
      int k = 4 * kk + 2 * half + v;
      a[v] = wdn(fl, k, l);       // A[n=l][k] = Wdn[k][n]
      b[v] = tp[k * 17 + l];      // B[k][c=l]
    }
    o = wmma4(a, b, o);
  }
#pragma unroll
  for (int vi = 0; vi < 8; ++vi) {
    int orow = r0 + vi + 8 * half;
    out[((size_t)nc * OUTS + orow) * OUTS + (j0 + l)] = o[vi];
  }
}

// ---------------------------------------------------------------------------
extern "C" void kernel_launch(void* const* d_in, const int* in_sizes, int n_in,
                              void* d_out, int out_size, void* d_ws,
                              size_t ws_size, hipStream_t stream) {
  (void)in_sizes; (void)n_in; (void)out_size; (void)ws_size;
  const float* img = (const float*)d_in[0];
  const float* th  = (const float*)d_in[1];
  const float* ls  = (const float*)d_in[2];
  const float* tx  = (const float*)d_in[3];
  const float* ty  = (const float*)d_in[4];
  const float* f   = (const float*)d_in[5];

  float* ws  = (float*)d_ws;
  float* aff = ws;          // 32*8 floats
  float* t2  = ws + 256;    // 96 * 664 * 664 floats (~169 MB intermediate)

  affine_prep<<<1, 32, 0, stream>>>(th, ls, tx, ty, aff);
  up2x<<<NIMG * NCH * 42 * 42, 32, 0, stream>>>(img, f, t2);
  warp_down<<<NIMG * NCH * 16 * 16, 32, 0, stream>>>(t2, f, aff,
                                                     (float*)d_out);
}